// AtomToFactor_6451040878620
// MI455X (gfx1250) — compile-verified
//
#include <hip/hip_runtime.h>

typedef __attribute__((ext_vector_type(2))) float v2f;
typedef __attribute__((ext_vector_type(8))) float v8f;

// One block = 128 threads = 4 wave32, handles 32 factor rows (two 16-row
// M-tiles). Wave w computes hidden columns [16w, 16w+16) for both tiles;
// each B (weight) fragment feeds two independent WMMAs. Layer3 runs on
// waves 0 and 1 (one M-tile each). All k-loops fully unrolled so the
// symmetrize-pass weight-row permutation and every load offset fold to
// compile-time immediates.
template<int K>
__global__ __launch_bounds__(128)
void factor_mlp(const float* __restrict__ atom_feat,
                const float* __restrict__ repr,
                const int*   __restrict__ idx,
                const float* __restrict__ W1, const float* __restrict__ b1,
                const float* __restrict__ W2, const float* __restrict__ b2,
                const float* __restrict__ W3, const float* __restrict__ b3,
                float* __restrict__ out)
{
    constexpr int DIN  = 64 * K + 1;    // k*ATOM_DIM + REPR
    constexpr int DPAD = 64 * K + 4;    // padded to multiple of WMMA K-step 4
    constexpr int NCH1 = DPAD / 4;      // k-chunks in layer 1 (last one peeled)
    constexpr int ROWS = 32;            // two 16-row M tiles per block

    __shared__ __align__(16) float Xs [ROWS * DPAD];
    __shared__ __align__(16) float H1s[ROWS * 64];
    __shared__ __align__(16) float H2s[ROWS * 64];

    const int base = blockIdx.x * ROWS;
    const int tid  = threadIdx.x;
    const int lane = tid & 31;
    const int wv   = tid >> 5;        // wave id 0..3
    const int half = lane >> 4;       // K-half for A/B fragments
    const int n    = lane & 15;       // M for A-frag, N for B/C/D
    const int wcol = wv * 16 + n;     // hidden column owned by this lane

    // ---- gather: 32 rows x K atoms x 64 feats, float4 granularity ----
    const float4* af4 = reinterpret_cast<const float4*>(atom_feat);
    float4* xs4 = reinterpret_cast<float4*>(Xs);
    for (int c = tid; c < ROWS * K * 16; c += 128) {
        int i   = c / (K * 16);
        int rem = c - i * (K * 16);
        int s   = rem >> 4;           // atom slot within factor
        int q   = rem & 15;           // float4 chunk within 64 feats
        int atom = idx[(base + i) * K + s];
        xs4[i * (DPAD / 4) + s * 16 + q] = af4[atom * 16 + q];
    }
    if (tid < ROWS) {                 // repr column + zero padding
        Xs[tid * DPAD + 64 * K + 0] = repr[base + tid];
        Xs[tid * DPAD + 64 * K + 1] = 0.f;
        Xs[tid * DPAD + 64 * K + 2] = 0.f;
        Xs[tid * DPAD + 64 * K + 3] = 0.f;
    }
    __syncthreads();

    // per-lane base pointers: all inner-loop offsets become immediates
    const float* aP0 = &Xs [n * DPAD + 2 * half];             // A tile 0
    const float* aP1 = aP0 + 16 * DPAD;                       // A tile 1
    const float* w1l = W1 + wcol + 128 * half;                // B rows (+2*half)
    const float* w2l = W2 + wcol + 128 * half;
    const float* h1P0 = &H1s[n * 64 + 2 * half];
    const float* h1P1 = h1P0 + 16 * 64;
    float* h1w = &H1s[(8 * half) * 64 + wcol];
    float* h2w = &H2s[(8 * half) * 64 + wcol];

    const float b1v = b1[wcol];
    const float b2v = b2[wcol];

    // layer-3 accumulator (waves 0/1, tile = wv); bias enters once per pass
    v8f acc;
    {
        float b3v = (n < 10) ? b3[n] : 0.f;
#pragma unroll
        for (int r = 0; r < 8; ++r) acc[r] = 2.f * b3v;
    }
    const float* h2P  = &H2s[(wv * 16 + n) * 64 + 2 * half];  // layer3 A base
    const float* w3l  = W3 + n + 20 * half;                   // (2*half)*10 + n

#pragma unroll
    for (int pass = 0; pass < 2; ++pass) {
        // ---- layer 1: X[32 x DPAD] @ W1[DIN x 64] ----
        v8f c0, c1;
#pragma unroll
        for (int r = 0; r < 8; ++r) { c0[r] = b1v; c1[r] = b1v; }

#pragma unroll
        for (int kk = 0; kk + 1 < NCH1; ++kk) {
            const int cb  = kk * 4;                       // cb < 64*K here
            // reversed pass: x_r @ W1 == x_f @ (slot-reversed-rows W1)
            const int cbs = pass ? (((K - 1 - (cb >> 6)) << 6) | (cb & 63)) : cb;
            v2f a0, a1, b;
            a0.x = aP0[cb]; a0.y = aP0[cb + 1];
            a1.x = aP1[cb]; a1.y = aP1[cb + 1];
            b.x  = w1l[cbs * 64];
            b.y  = w1l[cbs * 64 + 64];
            c0 = __builtin_amdgcn_wmma_f32_16x16x4_f32(false, a0, false, b, (short)0, c0, false, false);
            c1 = __builtin_amdgcn_wmma_f32_16x16x4_f32(false, a1, false, b, (short)0, c1, false, false);
        }
        {   // peeled final chunk: rows [64K, 64K+4); only row 64K (repr) is
            // real (same slot in both passes); padded X cols are zero.
            const int cb = 64 * K;
            v2f a0, a1, b;
            a0.x = aP0[cb]; a0.y = aP0[cb + 1];
            a1.x = aP1[cb]; a1.y = aP1[cb + 1];
            b.x  = half ? 0.f : W1[(64 * K) * 64 + wcol];
            b.y  = 0.f;
            c0 = __builtin_amdgcn_wmma_f32_16x16x4_f32(false, a0, false, b, (short)0, c0, false, false);
            c1 = __builtin_amdgcn_wmma_f32_16x16x4_f32(false, a1, false, b, (short)0, c1, false, false);
        }
#pragma unroll
        for (int r = 0; r < 8; ++r) {
            h1w[r * 64]            = fmaxf(c0[r], 0.f);
            h1w[(16 + r) * 64]     = fmaxf(c1[r], 0.f);
        }
        __syncthreads();

        // ---- layer 2: H1[32 x 64] @ W2[64 x 64] ----
#pragma unroll
        for (int r = 0; r < 8; ++r) { c0[r] = b2v; c1[r] = b2v; }
#pragma unroll
        for (int kk = 0; kk < 16; ++kk) {
            const int cb = kk * 4;
            v2f a0, a1, b;
            a0.x = h1P0[cb]; a0.y = h1P0[cb + 1];
            a1.x = h1P1[cb]; a1.y = h1P1[cb + 1];
            b.x  = w2l[cb * 64];
            b.y  = w2l[cb * 64 + 64];
            c0 = __builtin_amdgcn_wmma_f32_16x16x4_f32(false, a0, false, b, (short)0, c0, false, false);
            c1 = __builtin_amdgcn_wmma_f32_16x16x4_f32(false, a1, false, b, (short)0, c1, false, false);
        }
#pragma unroll
        for (int r = 0; r < 8; ++r) {
            h2w[r * 64]            = fmaxf(c0[r], 0.f);
            h2w[(16 + r) * 64]     = fmaxf(c1[r], 0.f);
        }
        __syncthreads();

        // ---- layer 3: H2[16 x 64] @ W3[64 x 10] (waves 0/1, tile = wv) ----
        if (wv < 2) {
#pragma unroll
            for (int kk = 0; kk < 16; ++kk) {
                const int cb = kk * 4;
                v2f a, b;
                a.x = h2P[cb]; a.y = h2P[cb + 1];
                b.x = (n < 10) ? w3l[cb * 10]      : 0.f;
                b.y = (n < 10) ? w3l[cb * 10 + 10] : 0.f;
                acc = __builtin_amdgcn_wmma_f32_16x16x4_f32(false, a, false, b, (short)0, acc, false, false);
            }
        }
        __syncthreads();   // protect H1/H2 before next pass overwrites
    }

    if (wv < 2 && n < 10) {
#pragma unroll
        for (int r = 0; r < 8; ++r)
            out[(base + wv * 16 + r + 8 * half) * 10 + n] = acc[r];
    }
}

extern "C" void kernel_launch(void* const* d_in, const int* in_sizes, int n_in,
                              void* d_out, int out_size, void* d_ws, size_t ws_size,
                              hipStream_t stream)
{
    const float* atom  = (const float*)d_in[0];
    const float* brepr = (const float*)d_in[1];
    const float* arepr = (const float*)d_in[2];
    const float* trepr = (const float*)d_in[3];
    const int*   bidx  = (const int*)d_in[4];
    const int*   aidx  = (const int*)d_in[5];
    const int*   tidx  = (const int*)d_in[6];
    const float* bp[6]; for (int i = 0; i < 6; ++i) bp[i] = (const float*)d_in[7 + i];
    const float* ap[6]; for (int i = 0; i < 6; ++i) ap[i] = (const float*)d_in[13 + i];
    const float* tp[6]; for (int i = 0; i < 6; ++i) tp[i] = (const float*)d_in[19 + i];
    float* out = (float*)d_out;

    // All row counts divide by 32 exactly: 6250 / 9375 / 15625 tiles.
    factor_mlp<2><<<200000 / 32, 128, 0, stream>>>(
        atom, brepr, bidx, bp[0], bp[1], bp[2], bp[3], bp[4], bp[5], out);
    factor_mlp<3><<<300000 / 32, 128, 0, stream>>>(
        atom, arepr, aidx, ap[0], ap[1], ap[2], ap[3], ap[4], ap[5], out + 200000 * 10);
    factor_mlp<4><<<500000 / 32, 128, 0, stream>>>(
        atom, trepr, tidx, tp[0], tp[1], tp[2], tp[3], tp[4], tp[5], out + 500000 * 10);
}